// OriginalGNNModelWithGAT_84224308674632
// MI455X (gfx1250) — compile-verified
//
#include <hip/hip_runtime.h>
#include <math.h>

// ---------------------------------------------------------------------------
// GAT (2-layer, PyG-style) for MI455X / gfx1250.
// Dense GEMMs use V_WMMA_F32_16X16X4_F32 (full f32 precision matrix pipe).
// Edge softmax/aggregation uses global f32/int atomics (L2-resident).
// ---------------------------------------------------------------------------

typedef __attribute__((ext_vector_type(2))) float v2f;
typedef __attribute__((ext_vector_type(8))) float v8f;

#define HEADS 4
#define CHANS 64          // per-head channels (both layers)
#define HC    256         // HEADS*CHANS

// ---------------- f32 WMMA GEMM: C[M,N] = A[M,K] @ B[K,N] (+ col bias) -----
// One wave computes one 16x16 tile of C. Grid sized EXACTLY: every wave has
// full EXEC (required by WMMA). M,N multiples of 16; K multiple of 4.
__global__ void gemm_f32_wmma(const float* __restrict__ A,
                              const float* __restrict__ B,
                              float* __restrict__ C,
                              const float* __restrict__ bias,
                              int N, int K) {
  const int wavesPerBlock = blockDim.x >> 5;
  const int wave = threadIdx.x >> 5;
  const int lane = threadIdx.x & 31;
  const int tile = blockIdx.x * wavesPerBlock + wave;
  const int tilesN = N >> 4;
  const int tm = tile / tilesN;
  const int tn = tile - tm * tilesN;

  const int row  = lane & 15;            // A row within tile / C col
  const int koff = (lane >> 4) << 1;     // lanes 16-31 hold K+2,K+3
  const int col  = lane & 15;

  const float* Arow = A + (size_t)(tm * 16 + row) * K + koff;
  const float* Bcol = B + (size_t)koff * N + tn * 16 + col;

  v8f acc = {0.f, 0.f, 0.f, 0.f, 0.f, 0.f, 0.f, 0.f};
  for (int k0 = 0; k0 < K; k0 += 4) {
    v2f a;
    a.x = Arow[k0];
    a.y = Arow[k0 + 1];
    v2f b;
    b.x = Bcol[(size_t)k0 * N];
    b.y = Bcol[(size_t)(k0 + 1) * N];
    acc = __builtin_amdgcn_wmma_f32_16x16x4_f32(false, a, false, b,
                                                (short)0, acc, false, false);
  }

  const float bb = bias ? bias[tn * 16 + col] : 0.f;
  const int rowBase = (lane >> 4) << 3;   // lanes 16-31 -> M+8
  float* Crow = C + (size_t)(tm * 16 + rowBase) * N + tn * 16 + col;
#pragma unroll
  for (int v = 0; v < 8; ++v)
    Crow[(size_t)v * N] = acc[v] + bb;
}

// ---------------- attention coefficients: a_s/a_d[n,h] = <h[n,h,:], att> ---
__global__ void attn_coef_kernel(const float* __restrict__ h,
                                 const float* __restrict__ att_s,
                                 const float* __restrict__ att_d,
                                 float* __restrict__ as_,
                                 float* __restrict__ ad_, int NH) {
  int idx = blockIdx.x * blockDim.x + threadIdx.x;
  if (idx >= NH) return;
  int n = idx >> 2, hh = idx & 3;
  const float* hp = h + (size_t)n * HC + hh * CHANS;
  const float* sp = att_s + hh * CHANS;
  const float* dp = att_d + hh * CHANS;
  float ss = 0.f, sd = 0.f;
#pragma unroll 8
  for (int c = 0; c < CHANS; ++c) {
    float v = hp[c];
    ss += v * sp[c];
    sd += v * dp[c];
  }
  as_[idx] = ss;
  ad_[idx] = sd;
}

// ---------------- init / zero ----------------------------------------------
__global__ void init_attn_state(float* __restrict__ emax,
                                float* __restrict__ denom, int NH) {
  int i = blockIdx.x * blockDim.x + threadIdx.x;
  if (i < NH) { emax[i] = -__builtin_inff(); denom[i] = 0.f; }
}

__global__ void zero_kernel(float* __restrict__ p, int n) {
  int i = blockIdx.x * blockDim.x + threadIdx.x;
  if (i < n) p[i] = 0.f;
}

// monotone float atomic-max via integer atomics (sign-aware bit trick)
__device__ inline void atomicMaxFloat(float* addr, float val) {
  if (val >= 0.f)
    atomicMax((int*)addr, __float_as_int(val));
  else
    atomicMin((unsigned int*)addr, __float_as_uint(val));
}

__device__ inline void edge_endpoints(const int* __restrict__ ei, int E_raw,
                                      int e, int& s, int& d) {
  if (e < E_raw) { s = ei[e]; d = ei[E_raw + e]; }
  else           { s = d = e - E_raw; }   // self-loops appended
}

// ---------------- edge pass 1: e = leaky_relu(a_s[src]+a_d[dst]); seg max --
__global__ void edge_max_kernel(const int* __restrict__ ei,
                                const float* __restrict__ as_,
                                const float* __restrict__ ad_,
                                float* __restrict__ ebuf,
                                float* __restrict__ emax,
                                int E_raw, int E_tot) {
  int idx = blockIdx.x * blockDim.x + threadIdx.x;
  if (idx >= E_tot * HEADS) return;
  int e = idx >> 2, hh = idx & 3;
  int s, d;
  edge_endpoints(ei, E_raw, e, s, d);
  float v = as_[s * HEADS + hh] + ad_[d * HEADS + hh];
  v = v > 0.f ? v : 0.2f * v;             // leaky_relu, NEG_SLOPE=0.2
  ebuf[idx] = v;
  atomicMaxFloat(&emax[d * HEADS + hh], v);
}

// ---------------- edge pass 2: ex = exp(e - emax[dst]); seg sum ------------
__global__ void edge_exp_kernel(const int* __restrict__ ei,
                                const float* __restrict__ emax,
                                float* __restrict__ ebuf,
                                float* __restrict__ denom,
                                int E_raw, int E_tot) {
  int idx = blockIdx.x * blockDim.x + threadIdx.x;
  if (idx >= E_tot * HEADS) return;
  int e = idx >> 2, hh = idx & 3;
  int d = (e < E_raw) ? ei[E_raw + e] : (e - E_raw);
  float x = expf(ebuf[idx] - emax[d * HEADS + hh]);
  ebuf[idx] = x;
  atomicAdd(&denom[d * HEADS + hh], x);
}

// ---------------- edge pass 3: out[dst] += alpha * h[src] ------------------
// one thread per (edge, channel); coalesced gather, f32 atomic scatter
__global__ void edge_agg_kernel(const int* __restrict__ ei,
                                const float* __restrict__ h,
                                const float* __restrict__ ebuf,
                                const float* __restrict__ denom,
                                float* __restrict__ out,
                                int E_raw, int E_tot) {
  int idx = blockIdx.x * blockDim.x + threadIdx.x;
  int e = idx >> 6;
  if (e >= E_tot) return;
  int c = idx & 63;
  int s, d;
  edge_endpoints(ei, E_raw, e, s, d);
  const float* hs = h + (size_t)s * HC + c;
  float* od = out + (size_t)d * HC + c;
#pragma unroll
  for (int hh = 0; hh < HEADS; ++hh) {
    float alpha = ebuf[e * HEADS + hh] / denom[d * HEADS + hh];
    atomicAdd(od + hh * CHANS, hs[hh * CHANS] * alpha);
  }
}

// ---------------- layer-1 epilogue: o = elu(o + b1) ------------------------
__global__ void bias_elu_kernel(float* __restrict__ o,
                                const float* __restrict__ b, int total) {
  int i = blockIdx.x * blockDim.x + threadIdx.x;
  if (i >= total) return;
  float v = o[i] + b[i & (HC - 1)];
  o[i] = v > 0.f ? v : (expf(v) - 1.f);   // ELU, alpha=1
}

// ---------------- layer-2 epilogue: m2 = mean_heads(o) + b2 ----------------
__global__ void mean_bias_kernel(const float* __restrict__ o,
                                 const float* __restrict__ b2,
                                 float* __restrict__ m2, int total) {
  int i = blockIdx.x * blockDim.x + threadIdx.x;
  if (i >= total) return;
  int n = i >> 6, c = i & 63;
  const float* p = o + (size_t)n * HC + c;
  m2[i] = 0.25f * (p[0] + p[CHANS] + p[2 * CHANS] + p[3 * CHANS]) + b2[c];
}

// ---------------------------------------------------------------------------
static inline void launch_gemm(const float* A, const float* B, float* C,
                               const float* bias, int M, int N, int K,
                               hipStream_t stream) {
  int tiles = (M / 16) * (N / 16);
  int wpb = (tiles % 4 == 0) ? 4 : (tiles % 2 == 0) ? 2 : 1;  // exact grid
  dim3 block(wpb * 32), grid(tiles / wpb);
  hipLaunchKernelGGL(gemm_f32_wmma, grid, block, 0, stream, A, B, C, bias, N, K);
}

extern "C" void kernel_launch(void* const* d_in, const int* in_sizes, int n_in,
                              void* d_out, int out_size, void* d_ws, size_t ws_size,
                              hipStream_t stream) {
  const float* x      = (const float*)d_in[0];
  const int*   ei     = (const int*)  d_in[1];
  const float* W1     = (const float*)d_in[2];
  const float* att_s1 = (const float*)d_in[3];
  const float* att_d1 = (const float*)d_in[4];
  const float* b1     = (const float*)d_in[5];
  const float* W2     = (const float*)d_in[6];
  const float* att_s2 = (const float*)d_in[7];
  const float* att_d2 = (const float*)d_in[8];
  const float* b2     = (const float*)d_in[9];
  const float* Wout   = (const float*)d_in[10];
  const float* bout   = (const float*)d_in[11];

  const int N     = in_sizes[0] / 128;     // 20000 nodes
  const int F     = 128;
  const int E_raw = in_sizes[1] / 2;       // 640000
  const int E_tot = E_raw + N;             // + self loops
  const int NH    = N * HEADS;

  // workspace layout (floats), buffers aliased across layers
  float* ws   = (float*)d_ws;
  size_t off  = 0;
  float* bufA = ws + off; off += (size_t)N * HC;      // h (layer features)
  float* bufB = ws + off; off += (size_t)N * HC;      // aggregation target
  float* m2   = ws + off; off += (size_t)N * CHANS;   // mean-head features
  float* asb  = ws + off; off += NH;
  float* adb  = ws + off; off += NH;
  float* emax = ws + off; off += NH;
  float* denom= ws + off; off += NH;
  float* ebuf = ws + off; off += (size_t)E_tot * HEADS;

  const int T = 256;
  dim3 blk(T);
  dim3 gNH((NH + T - 1) / T);
  dim3 gEH((E_tot * HEADS + T - 1) / T);
  dim3 gEC((E_tot * CHANS + T - 1) / T);
  dim3 gNC((N * HC + T - 1) / T);
  dim3 gM2((N * CHANS + T - 1) / T);

  // ---------------- Layer 1 ----------------
  launch_gemm(x, W1, bufA, nullptr, N, HC, F, stream);                 // h1 = x@W1
  hipLaunchKernelGGL(attn_coef_kernel, gNH, blk, 0, stream,
                     bufA, att_s1, att_d1, asb, adb, NH);
  hipLaunchKernelGGL(init_attn_state, gNH, blk, 0, stream, emax, denom, NH);
  hipLaunchKernelGGL(zero_kernel, gNC, blk, 0, stream, bufB, N * HC);
  hipLaunchKernelGGL(edge_max_kernel, gEH, blk, 0, stream,
                     ei, asb, adb, ebuf, emax, E_raw, E_tot);
  hipLaunchKernelGGL(edge_exp_kernel, gEH, blk, 0, stream,
                     ei, emax, ebuf, denom, E_raw, E_tot);
  hipLaunchKernelGGL(edge_agg_kernel, gEC, blk, 0, stream,
                     ei, bufA, ebuf, denom, bufB, E_raw, E_tot);
  hipLaunchKernelGGL(bias_elu_kernel, gNC, blk, 0, stream, bufB, b1, N * HC);

  // ---------------- Layer 2 ----------------
  launch_gemm(bufB, W2, bufA, nullptr, N, HC, HC, stream);             // h2 = o1@W2
  hipLaunchKernelGGL(attn_coef_kernel, gNH, blk, 0, stream,
                     bufA, att_s2, att_d2, asb, adb, NH);
  hipLaunchKernelGGL(init_attn_state, gNH, blk, 0, stream, emax, denom, NH);
  hipLaunchKernelGGL(zero_kernel, gNC, blk, 0, stream, bufB, N * HC);
  hipLaunchKernelGGL(edge_max_kernel, gEH, blk, 0, stream,
                     ei, asb, adb, ebuf, emax, E_raw, E_tot);
  hipLaunchKernelGGL(edge_exp_kernel, gEH, blk, 0, stream,
                     ei, emax, ebuf, denom, E_raw, E_tot);
  hipLaunchKernelGGL(edge_agg_kernel, gEC, blk, 0, stream,
                     ei, bufA, ebuf, denom, bufB, E_raw, E_tot);
  hipLaunchKernelGGL(mean_bias_kernel, gM2, blk, 0, stream,
                     bufB, b2, m2, N * CHANS);

  // ---------------- Classifier: out = m2 @ Wout + bout ----------------
  launch_gemm(m2, Wout, (float*)d_out, bout, N, 16, CHANS, stream);
}